// HodgkinHuxleyNeuron_40518721470627
// MI455X (gfx1250) — compile-verified
//
#include <hip/hip_runtime.h>
#include <hip/hip_bf16.h>

// Problem constants (match reference)
#define B_    16
#define T_    2000
#define N_    256
#define CHUNK_ 100                       // timesteps staged per LDS buffer
#define NCH_  (T_ / CHUNK_)              // 20 chunks
#define PLANE_ ((size_t)B_ * T_ * N_)    // 8,192,000 elements per output plane
#define DT_   0.01f

#define HAS_ASYNC_LDS __has_builtin(__builtin_amdgcn_global_load_async_to_lds_b128)

// v4i32 with explicit address-space-qualified pointee types, matching the
// builtin signature (v4i32 addrspace(1)*, v4i32 addrspace(3)*, imm, imm).
typedef int v4i __attribute__((ext_vector_type(4)));
typedef v4i __attribute__((address_space(1))) gbl_v4i;   // global-memory v4i32
typedef v4i __attribute__((address_space(3))) lds_v4i;   // LDS v4i32

__device__ __forceinline__ float fast_exp(float x) {
  // exp(x) = exp2(x * log2(e)) -> single v_exp_f32
  return __builtin_amdgcn_exp2f(x * 1.44269504088896340736f);
}

__device__ __forceinline__ void wait_async0() {
#if __has_builtin(__builtin_amdgcn_s_wait_asynccnt)
  __builtin_amdgcn_s_wait_asynccnt(0);
#else
  asm volatile("s_wait_asynccnt 0x0" ::: "memory");
#endif
}

__device__ __forceinline__ void hh_rates(float v, float& am, float& bm,
                                         float& ah, float& bh,
                                         float& an, float& bn) {
  // Six independent exponentials -> 6 parallel v_exp_f32 per step
  am = 0.1f * (v + 40.0f) / (1.0f - fast_exp(-(v + 40.0f) * 0.1f));
  bm = 4.0f * fast_exp(-(v + 65.0f) * (1.0f / 18.0f));
  ah = 0.07f * fast_exp(-(v + 65.0f) * 0.05f);
  bh = 1.0f / (1.0f + fast_exp(-(v + 35.0f) * 0.1f));
  an = 0.01f * (v + 55.0f) / (1.0f - fast_exp(-(v + 55.0f) * 0.1f));
  bn = 0.125f * fast_exp(-(v + 65.0f) * 0.0125f);
}

__device__ __forceinline__ float clip01(float x) {
  return fminf(fmaxf(x, 0.0f), 1.0f);
}

// Stage CHUNK_ x N_ floats (102,400 B) from global to LDS.
// 6400 b128 transfers; 256 threads x 25 iterations.
__device__ __forceinline__ void stage_chunk(const float* __restrict__ src,
                                            float* __restrict__ dstLds, int tid) {
#if HAS_ASYNC_LDS
  #pragma unroll
  for (int it = 0; it < (CHUNK_ * N_ / 4) / 256; ++it) {
    const int j = it * 256 + tid;  // float4 index
    __builtin_amdgcn_global_load_async_to_lds_b128(
        (gbl_v4i*)(src + j * 4),
        (lds_v4i*)(dstLds + j * 4),
        /*imm offset*/ 0, /*cpol*/ 0);
  }
#else
  #pragma unroll
  for (int it = 0; it < (CHUNK_ * N_ / 4) / 256; ++it) {
    const int j = it * 256 + tid;
    ((float4*)dstLds)[j] = ((const float4*)src)[j];
  }
#endif
}

extern "C" __global__ __launch_bounds__(256, 1)
void hh_fused_scan(const float* __restrict__ I, float* __restrict__ out) {
  extern __shared__ float smem[];            // 2 x CHUNK_ x N_ floats = 200 KB
  const int tid = threadIdx.x;               // neuron index 0..255
  const int b   = blockIdx.x;                // batch index 0..15

  float* buf0 = smem;
  float* buf1 = smem + CHUNK_ * N_;

  const float* Ib = I + (size_t)b * T_ * N_;

  float* spikes = out;
  float* vs     = out + PLANE_;
  float* ms     = out + 2 * PLANE_;
  float* hs     = out + 3 * PLANE_;
  float* ns     = out + 4 * PLANE_;

  // ---- initial state (reference: steady-state gates at v0 = -65) ----
  float v = -65.0f;
  float m, h, n;
  {
    float am, bm, ah, bh, an, bn;
    hh_rates(v, am, bm, ah, bh, an, bn);
    m = am / (am + bm);
    h = ah / (ah + bh);
    n = an / (an + bn);
  }
  float prev_v = 1.0f;  // >= 0 so spike at t=0 is 0 (matches zero padding)

  const size_t base = (size_t)b * T_ * N_ + (size_t)tid;

  // Prefetch chunk 0
  stage_chunk(Ib, buf0, tid);

  for (int c = 0; c < NCH_; ++c) {
#if HAS_ASYNC_LDS
    wait_async0();                 // chunk c DMA complete (this wave)
#endif
    __syncthreads();               // all waves' staging visible; prev buffer free

    float* rbuf = (c & 1) ? buf1 : buf0;
    if (c + 1 < NCH_) {
      stage_chunk(Ib + (size_t)(c + 1) * CHUNK_ * N_,
                  (c & 1) ? buf0 : buf1, tid);   // overlap DMA with compute
    }

    #pragma unroll 2
    for (int tl = 0; tl < CHUNK_; ++tl) {
      const float i_t = rbuf[tl * N_ + tid];     // bank-conflict-free

      // membrane currents with OLD state
      const float i_na = 120.0f * m * m * m * h * (v - 50.0f);
      const float i_k  = 36.0f * (n * n) * (n * n) * (v + 77.0f);
      const float i_l  = 0.3f * (v + 54.4f);
      const float dv   = (-i_na - i_k - i_l + i_t) * DT_;  // C_M = 1

      float am, bm, ah, bh, an, bn;
      hh_rates(v, am, bm, ah, bh, an, bn);

      const float m_new = clip01(m + (am * (1.0f - m) - bm * m) * DT_);
      const float h_new = clip01(h + (ah * (1.0f - h) - bh * h) * DT_);
      const float n_new = clip01(n + (an * (1.0f - n) - bn * n) * DT_);
      const float v_new = v + dv;

      const float spike = (v_new >= 0.0f && prev_v < 0.0f) ? 1.0f : 0.0f;

      const size_t idx = base + (size_t)(c * CHUNK_ + tl) * N_;
      __builtin_nontemporal_store(spike, spikes + idx);
      __builtin_nontemporal_store(v_new, vs + idx);
      __builtin_nontemporal_store(m_new, ms + idx);
      __builtin_nontemporal_store(h_new, hs + idx);
      __builtin_nontemporal_store(n_new, ns + idx);

      prev_v = v_new;
      v = v_new; m = m_new; h = h_new; n = n_new;
    }
  }
}

extern "C" void kernel_launch(void* const* d_in, const int* in_sizes, int n_in,
                              void* d_out, int out_size, void* d_ws, size_t ws_size,
                              hipStream_t stream) {
  (void)in_sizes; (void)n_in; (void)out_size; (void)d_ws; (void)ws_size;
  const float* I = (const float*)d_in[0];
  float* out = (float*)d_out;

  const size_t shbytes = (size_t)2 * CHUNK_ * N_ * sizeof(float);  // 204,800 B
  // Allow >64KB dynamic LDS (WGP has 320KB on CDNA5)
  (void)hipFuncSetAttribute((const void*)hh_fused_scan,
                            hipFuncAttributeMaxDynamicSharedMemorySize,
                            (int)shbytes);

  hipLaunchKernelGGL(hh_fused_scan, dim3(B_), dim3(N_), shbytes, stream, I, out);
}